// TwinSAGE_53180285059698
// MI455X (gfx1250) — compile-verified
//
#include <hip/hip_runtime.h>
#include <hip/hip_bf16.h>

// ---------------------------------------------------------------------------
// TwinSAGE on MI455X (gfx1250, wave32).
// GEMMs run on V_WMMA_F32_16X16X4_F32 (full fp32 accuracy, matrix pipe).
// Edge aggregation uses no-return f32 global atomics (L2-resident targets).
// ---------------------------------------------------------------------------

typedef __attribute__((ext_vector_type(2))) float v2f;
typedef __attribute__((ext_vector_type(8))) float v8f;

#define N0c  500000
#define N1c  100000
#define Bc   10000
#define E0c  1000000
#define E1c  100000
#define FINc 128
#define HIDc 256
#define Cc   40

// ---------------------------------------------------------------------------
__global__ void zero_f32(float* __restrict__ p, int n) {
  int i = blockIdx.x * blockDim.x + threadIdx.x;
  int stride = gridDim.x * blockDim.x;
  for (; i < n; i += stride) p[i] = 0.0f;
}

// ---------------------------------------------------------------------------
// One thread per (edge, feature). F = 1<<logF (128 or 256).
// sums[dst] += feat[src]; cnt[dst] += 1 (feature-0 thread only).
__global__ void scatter_edges(const float* __restrict__ feat,
                              const int* __restrict__ src,
                              const int* __restrict__ dst,
                              float* __restrict__ sums,
                              float* __restrict__ cnt,
                              int nEdges, int logF) {
  int idx = blockIdx.x * blockDim.x + threadIdx.x;
  int total = nEdges << logF;
  if (idx >= total) return;
  int e = idx >> logF;
  int f = idx & ((1 << logF) - 1);
  int s = src[e];
  int d = dst[e];
  float v = feat[((long long)s << logF) + f];
  __hip_atomic_fetch_add(&sums[((long long)d << logF) + f], v,
                         __ATOMIC_RELAXED, __HIP_MEMORY_SCOPE_AGENT);
  if (f == 0) {
    __hip_atomic_fetch_add(&cnt[d], 1.0f,
                           __ATOMIC_RELAXED, __HIP_MEMORY_SCOPE_AGENT);
  }
}

// ---------------------------------------------------------------------------
// out[rows RT*16*blockIdx.x .. +RT*16, cols 0..255] =
//   relu( (Asum/max(cnt,1)) @ Wl  +  Aself @ Wr  +  bias )
// HAS_AGG=false -> only the self term. bias==nullptr -> 0.
// Block: 256 threads = 8 waves. Each wave owns 2 column tiles (wave, wave+8)
// and RT row tiles -> 2*RT accumulators, 2*RT (x2 if agg) WMMAs per k-step.
// A fragments feed both column tiles; B fragments feed all RT row tiles.
// K = 1<<logK (128 or 256). N fixed at HIDc=256. EXEC all-ones throughout.
template <bool HAS_AGG, int RT>
__global__ __launch_bounds__(256)
void sage_gemm(const float* __restrict__ Asum, const float* __restrict__ cnt,
               const float* __restrict__ Aself,
               const float* __restrict__ Wl, const float* __restrict__ Wr,
               const float* __restrict__ bias,
               float* __restrict__ out, int logK) {
  const int K = 1 << logK;
  const int LDK = 260;                          // stride%64==4 -> conflict-free
  __shared__ float ldsSelf[16 * RT * LDK];
  __shared__ float ldsAgg[HAS_AGG ? 16 * RT * LDK : 1];
  const int t = threadIdx.x;
  const long long rowBase = (long long)blockIdx.x * (16 * RT);

  // Cooperative stage of the (16*RT)xK A tiles (mean-normalize agg here).
  for (int i = t; i < ((16 * RT) << logK); i += 256) {
    int r = i >> logK;
    int c = i & (K - 1);
    long long g = ((rowBase + r) << logK) + c;
    ldsSelf[r * LDK + c] = Aself[g];
    if (HAS_AGG) {
      float inv = 1.0f / fmaxf(cnt[rowBase + r], 1.0f);
      ldsAgg[r * LDK + c] = Asum[g] * inv;
    }
  }
  __syncthreads();

  const int wave = t >> 5;                      // 0..7
  const int lane = t & 31;
  const int rowA = lane & 15;                   // A fragment: M index
  const int kb   = (lane >> 4) << 1;            // A/B fragment: K sub-offset
  const int nc0  = (wave << 4) + (lane & 15);   // column tile 0
  const int nc1  = nc0 + 128;                   // column tile 1 (wave+8)

  v8f acc[RT][2];
  {
    float b0 = (bias != nullptr) ? bias[nc0] : 0.0f;  // C depends on N only
    float b1 = (bias != nullptr) ? bias[nc1] : 0.0f;
    for (int rt = 0; rt < RT; ++rt)
      for (int v = 0; v < 8; ++v) { acc[rt][0][v] = b0; acc[rt][1][v] = b1; }
  }
  const float* wr0 = Wr + nc0;
  const float* wr1 = Wr + nc1;
  const float* wl0 = HAS_AGG ? (Wl + nc0) : nullptr;
  const float* wl1 = HAS_AGG ? (Wl + nc1) : nullptr;

  for (int k0 = 0; k0 < K; k0 += 4) {
    const int k = k0 + kb;
    // B fragments (one per weight matrix per column tile)
    v2f bR0, bR1;
    bR0.x = wr0[k * HIDc]; bR0.y = wr0[(k + 1) * HIDc];
    bR1.x = wr1[k * HIDc]; bR1.y = wr1[(k + 1) * HIDc];
    // A fragments (one per row tile per A matrix)
    v2f aS[RT];
    for (int rt = 0; rt < RT; ++rt) {
      aS[rt].x = ldsSelf[(rt * 16 + rowA) * LDK + k];
      aS[rt].y = ldsSelf[(rt * 16 + rowA) * LDK + k + 1];
    }
    if (HAS_AGG) {
      v2f bL0, bL1;
      bL0.x = wl0[k * HIDc]; bL0.y = wl0[(k + 1) * HIDc];
      bL1.x = wl1[k * HIDc]; bL1.y = wl1[(k + 1) * HIDc];
      v2f aG[RT];
      for (int rt = 0; rt < RT; ++rt) {
        aG[rt].x = ldsAgg[(rt * 16 + rowA) * LDK + k];
        aG[rt].y = ldsAgg[(rt * 16 + rowA) * LDK + k + 1];
      }
      for (int rt = 0; rt < RT; ++rt) {
        acc[rt][0] = __builtin_amdgcn_wmma_f32_16x16x4_f32(
            false, aS[rt], false, bR0, (short)0, acc[rt][0], false, false);
        acc[rt][1] = __builtin_amdgcn_wmma_f32_16x16x4_f32(
            false, aS[rt], false, bR1, (short)0, acc[rt][1], false, false);
        acc[rt][0] = __builtin_amdgcn_wmma_f32_16x16x4_f32(
            false, aG[rt], false, bL0, (short)0, acc[rt][0], false, false);
        acc[rt][1] = __builtin_amdgcn_wmma_f32_16x16x4_f32(
            false, aG[rt], false, bL1, (short)0, acc[rt][1], false, false);
      }
    } else {
      for (int rt = 0; rt < RT; ++rt) {
        acc[rt][0] = __builtin_amdgcn_wmma_f32_16x16x4_f32(
            false, aS[rt], false, bR0, (short)0, acc[rt][0], false, false);
        acc[rt][1] = __builtin_amdgcn_wmma_f32_16x16x4_f32(
            false, aS[rt], false, bR1, (short)0, acc[rt][1], false, false);
      }
    }
  }

  // ReLU + store. C/D layout: VGPR v holds M=v (lanes 0-15), M=v+8 (16-31).
  const int mhi = (lane >> 4) * 8;
  for (int rt = 0; rt < RT; ++rt) {
    for (int v = 0; v < 8; ++v) {
      long long m = rowBase + rt * 16 + mhi + v;
      out[m * (long long)HIDc + nc0] = fmaxf(acc[rt][0][v], 0.0f);
      out[m * (long long)HIDc + nc1] = fmaxf(acc[rt][1][v], 0.0f);
    }
  }
}

// ---------------------------------------------------------------------------
// Per-row: scores = [h.hq, h2.h2q]; alpha = softmax; mix = a0*h + a1*h2;
// out = mix @ Wout + bout.  One 256-thread block per row.
__global__ __launch_bounds__(256)
void attn_out(const float* __restrict__ h,  const float* __restrict__ hq,
              const float* __restrict__ h2, const float* __restrict__ h2q,
              const float* __restrict__ Wout, const float* __restrict__ bout,
              float* __restrict__ out) {
  __shared__ float r0[256], r1[256], mix[256];
  const int row = blockIdx.x;
  const int t = threadIdx.x;
  const long long base = (long long)row * HIDc + t;
  float hv = h[base], h2v = h2[base];
  r0[t] = hv * hq[base];
  r1[t] = h2v * h2q[base];
  __syncthreads();
  for (int s = 128; s > 0; s >>= 1) {
    if (t < s) { r0[t] += r0[t + s]; r1[t] += r1[t + s]; }
    __syncthreads();
  }
  float s0 = r0[0], s1 = r1[0];           // TEMP == 1.0
  float m = fmaxf(s0, s1);
  float e0 = __expf(s0 - m), e1 = __expf(s1 - m);
  float inv = 1.0f / (e0 + e1);
  mix[t] = (e0 * inv) * hv + (e1 * inv) * h2v;
  __syncthreads();
  if (t < Cc) {
    float acc = bout[t];
    for (int k = 0; k < HIDc; ++k) acc += mix[k] * Wout[k * Cc + t];
    out[(long long)row * Cc + t] = acc;
  }
}

// ---------------------------------------------------------------------------
extern "C" void kernel_launch(void* const* d_in, const int* in_sizes, int n_in,
                              void* d_out, int out_size, void* d_ws, size_t ws_size,
                              hipStream_t stream) {
  const float* x    = (const float*)d_in[0];
  const int*   src0 = (const int*)d_in[1];
  const int*   dst0 = (const int*)d_in[2];
  const int*   src1 = (const int*)d_in[3];
  const int*   dst1 = (const int*)d_in[4];
  const float* W1l  = (const float*)d_in[5];
  const float* b1   = (const float*)d_in[6];
  const float* W1r  = (const float*)d_in[7];
  const float* W2l  = (const float*)d_in[8];
  const float* b2   = (const float*)d_in[9];
  const float* W2r  = (const float*)d_in[10];
  const float* Wout = (const float*)d_in[11];
  const float* bout = (const float*)d_in[12];
  float* out = (float*)d_out;

  // Workspace layout (floats). Zero-initialized region first.
  float* ws = (float*)d_ws;
  size_t off = 0;
  float* s0   = ws + off; off += (size_t)N1c * FINc;   // 12.8M
  float* cnt0 = ws + off; off += (size_t)N1c;          // 100k
  float* s1   = ws + off; off += (size_t)Bc * HIDc;    // 2.56M
  float* cnt1 = ws + off; off += (size_t)Bc;           // 10k
  const int zeroN = (int)off;
  float* h    = ws + off; off += (size_t)N1c * HIDc;   // 25.6M
  float* hq   = ws + off; off += (size_t)Bc * HIDc;    // h_  (query path L1)
  float* h2   = ws + off; off += (size_t)Bc * HIDc;
  float* h2q  = ws + off; off += (size_t)Bc * HIDc;    // h2_ (query path L2)

  // 1) zero segment sums + counts
  zero_f32<<<4096, 256, 0, stream>>>(ws, zeroN);

  // 2) layer-1 edge aggregation: s0[dst0] += x[src0]; cnt0[dst0]++
  scatter_edges<<<(E0c * FINc) / 256, 256, 0, stream>>>(x, src0, dst0, s0, cnt0,
                                                        E0c, 7);

  // 3) h = relu(mean(s0) @ W1l + b1 + x[:N1] @ W1r)   (WMMA, K=128, 32-row blk)
  sage_gemm<true, 2><<<N1c / 32, 256, 0, stream>>>(s0, cnt0, x, W1l, W1r, b1,
                                                   h, 7);

  // 4) hq = relu(x[:B] @ W1r)                         (WMMA, K=128)
  sage_gemm<false, 1><<<Bc / 16, 256, 0, stream>>>(nullptr, nullptr, x,
                                                   nullptr, W1r, nullptr, hq, 7);

  // 5) layer-2 edge aggregation over h: s1[dst1] += h[src1]; cnt1[dst1]++
  scatter_edges<<<(E1c * HIDc) / 256, 256, 0, stream>>>(h, src1, dst1, s1, cnt1,
                                                        E1c, 8);

  // 6) h2 = relu(mean(s1) @ W2l + b2 + h[:B] @ W2r)   (WMMA, K=256)
  sage_gemm<true, 1><<<Bc / 16, 256, 0, stream>>>(s1, cnt1, h, W2l, W2r, b2,
                                                  h2, 8);

  // 7) h2q = relu(hq @ W2r)                           (WMMA, K=256)
  sage_gemm<false, 1><<<Bc / 16, 256, 0, stream>>>(nullptr, nullptr, hq,
                                                   nullptr, W2r, nullptr, h2q, 8);

  // 8) attention mix + output projection
  attn_out<<<Bc, 256, 0, stream>>>(h, hq, h2, h2q, Wout, bout, out);
}